// AttentionSink_64836826300756
// MI455X (gfx1250) — compile-verified
//
#include <hip/hip_runtime.h>

// AttentionSink additive mask, computed analytically (never materialized).
// Memory-bound streaming kernel: b128 NT loads/stores, 2 vec4 per thread.

typedef float v4f __attribute__((ext_vector_type(4)));

namespace {

constexpr unsigned kSink   = 4;       // SINK_SIZE
constexpr int      kLeft   = 25;      // LEFT_CONTEXT (RIGHT_CONTEXT == 0)
constexpr float    kNegInf = -1.0e30f;
constexpr unsigned kBlock  = 256;     // 8 wave32s per block

__device__ __forceinline__ float mask_add(float x, unsigned i, unsigned j) {
  // allowed = (j < SINK) || (j >= i - LEFT && j <= i)
  const bool allowed = (j < kSink) || (((int)j >= (int)i - kLeft) && (j <= i));
  // Match reference bit-exactly: it adds 0.0f in the allowed case too.
  return x + (allowed ? 0.0f : kNegInf);
}

__global__ __launch_bounds__(kBlock) void sink_mask_vec4(
    const v4f* __restrict__ in, v4f* __restrict__ out,
    const int* __restrict__ seq_len_ptr, unsigned nvec)
{
  const unsigned tid    = blockIdx.x * kBlock + threadIdx.x;
  const unsigned stride = gridDim.x * kBlock;     // first/second half split
  const unsigned S      = (unsigned)*seq_len_ptr; // uniform s_load
  const bool     pow2   = (S & (S - 1u)) == 0u;   // true for S = 2048

  const unsigned idx0 = tid;
  const unsigned idx1 = tid + stride;

  // Issue both 128-bit NT loads up front (2 outstanding per thread).
  v4f a = (v4f)(0.0f);
  v4f b = (v4f)(0.0f);
  if (idx0 < nvec) a = __builtin_nontemporal_load(&in[idx0]);
  if (idx1 < nvec) b = __builtin_nontemporal_load(&in[idx1]);

  if (pow2) {
    const unsigned log2S  = 31u - (unsigned)__clz((int)S);
    const unsigned vshift = log2S - 2u;            // log2(S/4) vec4s per row
    const unsigned vmask  = (1u << vshift) - 1u;
    const unsigned smask  = S - 1u;

    if (idx0 < nvec) {
      const unsigned i  = (idx0 >> vshift) & smask;
      const unsigned j0 = (idx0 & vmask) << 2;
      a.x = mask_add(a.x, i, j0 + 0u);
      a.y = mask_add(a.y, i, j0 + 1u);
      a.z = mask_add(a.z, i, j0 + 2u);
      a.w = mask_add(a.w, i, j0 + 3u);
      __builtin_nontemporal_store(a, &out[idx0]);
    }
    if (idx1 < nvec) {
      const unsigned i  = (idx1 >> vshift) & smask;
      const unsigned j0 = (idx1 & vmask) << 2;
      b.x = mask_add(b.x, i, j0 + 0u);
      b.y = mask_add(b.y, i, j0 + 1u);
      b.z = mask_add(b.z, i, j0 + 2u);
      b.w = mask_add(b.w, i, j0 + 3u);
      __builtin_nontemporal_store(b, &out[idx1]);
    }
  } else {
    // Generic (uniform-dead for the reference shapes): per-element div/mod.
    if (idx0 < nvec) {
      const unsigned e = idx0 << 2;
      a.x = mask_add(a.x, ((e + 0u) / S) % S, (e + 0u) % S);
      a.y = mask_add(a.y, ((e + 1u) / S) % S, (e + 1u) % S);
      a.z = mask_add(a.z, ((e + 2u) / S) % S, (e + 2u) % S);
      a.w = mask_add(a.w, ((e + 3u) / S) % S, (e + 3u) % S);
      __builtin_nontemporal_store(a, &out[idx0]);
    }
    if (idx1 < nvec) {
      const unsigned e = idx1 << 2;
      b.x = mask_add(b.x, ((e + 0u) / S) % S, (e + 0u) % S);
      b.y = mask_add(b.y, ((e + 1u) / S) % S, (e + 1u) % S);
      b.z = mask_add(b.z, ((e + 2u) / S) % S, (e + 2u) % S);
      b.w = mask_add(b.w, ((e + 3u) / S) % S, (e + 3u) % S);
      __builtin_nontemporal_store(b, &out[idx1]);
    }
  }
}

// Scalar tail for out_size % 4 != 0 (not hit for the reference shapes).
__global__ __launch_bounds__(64) void sink_mask_tail(
    const float* __restrict__ in, float* __restrict__ out,
    const int* __restrict__ seq_len_ptr, unsigned start, unsigned n_total)
{
  const unsigned e = start + blockIdx.x * 64u + threadIdx.x;
  if (e >= n_total) return;
  const unsigned S = (unsigned)*seq_len_ptr;
  const unsigned i = (e / S) % S;
  const unsigned j = e % S;
  out[e] = mask_add(in[e], i, j);
}

} // namespace

extern "C" void kernel_launch(void* const* d_in, const int* in_sizes, int n_in,
                              void* d_out, int out_size, void* d_ws, size_t ws_size,
                              hipStream_t stream)
{
  (void)in_sizes; (void)n_in; (void)d_ws; (void)ws_size;

  const float* in   = reinterpret_cast<const float*>(d_in[0]);
  const int*   slen = reinterpret_cast<const int*>(d_in[1]);  // device scalar
  float*       out  = reinterpret_cast<float*>(d_out);

  const unsigned n    = (unsigned)out_size;
  const unsigned nvec = n / 4u;

  if (nvec) {
    // 2 float4s (32 B) per thread; halves are stride-split so every load
    // instruction is contiguous across the wave.
    const unsigned per_block = 2u * kBlock * 4u;  // elements per block
    const unsigned blocks    = (4u * nvec + per_block - 1u) / per_block;
    sink_mask_vec4<<<dim3(blocks), dim3(kBlock), 0, stream>>>(
        reinterpret_cast<const v4f*>(in), reinterpret_cast<v4f*>(out),
        slen, nvec);
  }

  const unsigned rem = n % 4u;
  if (rem) {
    sink_mask_tail<<<dim3(1), dim3(64), 0, stream>>>(in, out, slen, n - rem, n);
  }
}